// Affinity_GAT_34471407518282
// MI455X (gfx1250) — compile-verified
//
#include <hip/hip_runtime.h>
#include <cmath>

// ---------------------------------------------------------------------------
// GATv2 x2 for MI455X (gfx1250, wave32).
//  - Dense transforms: V_WMMA_F32_16X16X4_F32, B-panel staged in LDS and
//    shared by 8 waves per block (one 16-wide N column tile per block).
//  - Graph softmax-aggregation: wave-per-(edge,head), lane-parallel channels,
//    L2-resident atomics (encoded u32 max, f32 add).
// ---------------------------------------------------------------------------

typedef __attribute__((ext_vector_type(2))) float v2f;
typedef __attribute__((ext_vector_type(8))) float v8f;

#define LRELU_SLOPE 0.2f
#define GEMM_WAVES  8           // waves (row tiles) per block
#define KMAX        192         // max K over both layers

// ------------------------------ fill kernels -------------------------------
__global__ void fill_f32_kernel(float* __restrict__ p, float v, long n) {
  long i = (long)blockIdx.x * blockDim.x + threadIdx.x;
  if (i < n) p[i] = v;
}
__global__ void fill_u32_kernel(unsigned* __restrict__ p, unsigned v, long n) {
  long i = (long)blockIdx.x * blockDim.x + threadIdx.x;
  if (i < n) p[i] = v;
}

// ------------------------- WMMA f32 GEMM + bias ----------------------------
// C[M,N] = A[M,K] @ W[K,N] + bias[N].  Row-major.  M%16==0, N%16==0, K%4==0,
// K <= KMAX.  Block = 256 threads = 8 waves; all waves share one 16-wide N
// column tile (B panel staged once in LDS) and cover 8 consecutive 16-row M
// tiles.  K-loop of v_wmma_f32_16x16x4_f32.
//
// A-frag (16x4 f32, ISA 7.12.2): lane l<16 -> row m0+l, K=k+0/k+1 (v0/v1);
// lane l>=16 -> same rows, K=k+2/k+3.  Contiguous float2 at arow[k+2*half].
// B-frag (4x16): VGPR0 lanes0-15 -> K=k+0, lanes16-31 -> K=k+2; VGPR1 ->
// K=k+1 / K=k+3; column = n0 + (lane&15).
// C/D (16x16): VGPR j, lanes0-15 -> row m0+j, lanes16-31 -> row m0+j+8.
__global__ void gemm_bias_wmma(const float* __restrict__ A,
                               const float* __restrict__ W,
                               const float* __restrict__ bias,
                               float* __restrict__ C,
                               int M, int N, int K) {
  __shared__ float ldsB[KMAX * 16];          // 12 KB max (of 320 KB WGP LDS)

  const int tilesN  = N >> 4;
  const int nTile   = blockIdx.x % tilesN;
  const int mBlock  = blockIdx.x / tilesN;
  const int n0      = nTile << 4;
  const int waveId  = threadIdx.x >> 5;
  const int lane    = threadIdx.x & 31;
  const int half    = lane >> 4;             // 0: K pair {0,1}; 1: K pair {2,3}
  const int ln      = lane & 15;

  // ---- cooperative B-panel stage: W[0:K, n0:n0+16] -> LDS (row-major x16)
  for (int idx = threadIdx.x; idx < K * 16; idx += blockDim.x)
    ldsB[idx] = W[(size_t)(idx >> 4) * N + n0 + (idx & 15)];
  __syncthreads();

  const int m0 = (mBlock * GEMM_WAVES + waveId) << 4;
  if (m0 + 16 > M) return;

  const float* arow = A + (size_t)(m0 + ln) * K + 2 * half;   // 8B aligned
  const float* bp   = ldsB + ln;

  v8f acc = {};
  for (int k = 0; k < K; k += 4) {
    v2f a = *(const v2f*)(arow + k);          // K=k+2h, k+2h+1, row m0+ln
    const int kb = k + 2 * half;
    v2f b;
    b.x = bp[kb * 16];                        // K=kb,   col n0+ln  (LDS)
    b.y = bp[(kb + 1) * 16];                  // K=kb+1, col n0+ln  (LDS)
    acc = __builtin_amdgcn_wmma_f32_16x16x4_f32(
        /*neg_a=*/false, a, /*neg_b=*/false, b,
        /*c_mod=*/(short)0, acc, /*reuse_a=*/false, /*reuse_b=*/false);
  }

  const float bv = bias[n0 + ln];
  float* crow = C + (size_t)(m0 + 8 * half) * N + n0 + ln;
#pragma unroll
  for (int j = 0; j < 8; ++j)
    crow[(size_t)j * N] = acc[j] + bv;
}

// ------------------------- monotonic float encode --------------------------
// Order-preserving map float -> uint so segment-max can use atomic u32 max.
__device__ __forceinline__ unsigned enc_f32(float f) {
  unsigned u = __float_as_uint(f);
  return (u & 0x80000000u) ? ~u : (u | 0x80000000u);
}

// ------------------------- per-(edge,head) logits --------------------------
// One wave per (e,h); lanes cover channels c = lane, lane+32 (C%32==0).
// logit[e,h] = sum_c lrelu(xl[s,h,c] + xr[d,h,c] + ea[e]*We[h,c]) * att[h,c]
// then wave32 shuffle-reduce; lane 0 stores + encoded atomic max into mxenc.
__global__ void edge_logits_kernel(const float* __restrict__ xl,
                                   const float* __restrict__ xr,
                                   const float* __restrict__ We,
                                   const float* __restrict__ att,
                                   const long long* __restrict__ src,
                                   const long long* __restrict__ dst,
                                   const float* __restrict__ ea,
                                   float* __restrict__ logits,
                                   unsigned* __restrict__ mxenc,
                                   int E_, int H, int C) {
  const int wid  = blockIdx.x * (blockDim.x >> 5) + (threadIdx.x >> 5);
  const int lane = threadIdx.x & 31;
  if (wid >= E_ * H) return;
  const int e = wid / H;
  const int h = wid - e * H;
  const long long s = src[e];
  const long long d = dst[e];
  const float eav = ea[e];
  const int HC = H * C;

  const float* pl = xl + (size_t)s * HC + h * C;
  const float* pr = xr + (size_t)d * HC + h * C;
  const float* pw = We + h * C;
  const float* pa = att + h * C;

  float part = 0.f;
#pragma unroll
  for (int c = lane; c < C; c += 32) {        // coalesced across the wave
    float v = pl[c] + pr[c] + eav * pw[c];
    v = (v > 0.f) ? v : LRELU_SLOPE * v;
    part += v * pa[c];
  }
#pragma unroll
  for (int off = 16; off > 0; off >>= 1)      // wave32 reduction
    part += __shfl_xor(part, off, 32);

  if (lane == 0) {
    logits[wid] = part;
    atomicMax(&mxenc[(size_t)d * H + h], enc_f32(part));
  }
}

// ----------------- decode segment max; zero non-finite; den=0 --------------
__global__ void fix_max_kernel(const unsigned* __restrict__ mxenc,
                               float* __restrict__ mx,
                               float* __restrict__ den, long n) {
  long i = (long)blockIdx.x * blockDim.x + threadIdx.x;
  if (i >= n) return;
  const unsigned enc = mxenc[i];
  float f = 0.f;
  if (enc != 0u) {
    unsigned u = (enc & 0x80000000u) ? (enc & 0x7FFFFFFFu) : ~enc;
    f = __uint_as_float(u);
    if ((u & 0x7F800000u) == 0x7F800000u) f = 0.f;   // inf/nan -> 0
  }
  mx[i] = f;
  den[i] = 0.f;
}

// --------------------- exp(logit - max) and denominator --------------------
__global__ void edge_exp_kernel(const long long* __restrict__ dst,
                                float* __restrict__ logits,    // overwritten w/ ex
                                const float* __restrict__ mx,
                                float* __restrict__ den,
                                int E_, int H) {
  const int idx = blockIdx.x * blockDim.x + threadIdx.x;
  if (idx >= E_ * H) return;
  const int e = idx / H;
  const int h = idx - e * H;
  const long long d = dst[e];
  const float ex = expf(logits[idx] - mx[(size_t)d * H + h]);
  logits[idx] = ex;
  atomicAdd(&den[(size_t)d * H + h], ex);
}

// ----------------------- weighted scatter aggregation ----------------------
// One wave per (e,h); lanes cover channels c = lane, lane+32, so the f32
// atomic adds are lane-contiguous (dense cachelines at the L2 atomic units).
// out[d, h*C + c] += (ex/den) * xl[s, h*C + c]
__global__ void edge_scatter_kernel(const long long* __restrict__ src,
                                    const long long* __restrict__ dst,
                                    const float* __restrict__ ex,
                                    const float* __restrict__ den,
                                    const float* __restrict__ xl,
                                    float* __restrict__ out,
                                    int E_, int H, int C) {
  const int wid  = blockIdx.x * (blockDim.x >> 5) + (threadIdx.x >> 5);
  const int lane = threadIdx.x & 31;
  if (wid >= E_ * H) return;
  const int e = wid / H;
  const int h = wid - e * H;
  const long long s = src[e];
  const long long d = dst[e];
  const float alpha = ex[wid] / (den[(size_t)d * H + h] + 1e-16f);
  const int HC = H * C;
  const float* pl = xl + (size_t)s * HC + h * C;
  float* po = out + (size_t)d * HC + h * C;
#pragma unroll
  for (int c = lane; c < C; c += 32)
    atomicAdd(&po[c], alpha * pl[c]);
}

// ------------------------ bias + ELU (in place) ----------------------------
__global__ void finalize_elu_kernel(float* __restrict__ acc,
                                    const float* __restrict__ bias,
                                    long n, int HC) {
  long i = (long)blockIdx.x * blockDim.x + threadIdx.x;
  if (i >= n) return;
  float v = acc[i] + bias[i % HC];
  acc[i] = (v > 0.f) ? v : (expf(v) - 1.f);
}

// ---------------------------------------------------------------------------
static inline int cdiv(long a, long b) { return (int)((a + b - 1) / b); }

extern "C" void kernel_launch(void* const* d_in, const int* in_sizes, int n_in,
                              void* d_out, int out_size, void* d_ws, size_t ws_size,
                              hipStream_t stream) {
  (void)in_sizes; (void)n_in; (void)out_size; (void)ws_size;

  const int N = 50000, E = 800000, FIN = 128, HID = 64, H1 = 3, OUT = 64;
  const int HC1 = H1 * HID;   // 192

  // ------- inputs (setup_inputs order) -------
  const float*     x    = (const float*)d_in[0];
  const long long* ei   = (const long long*)d_in[1];
  const float*     ea   = (const float*)d_in[2];
  const float*     Wl1  = (const float*)d_in[3];
  const float*     bl1  = (const float*)d_in[4];
  const float*     Wr1  = (const float*)d_in[5];
  const float*     br1  = (const float*)d_in[6];
  const float*     We1  = (const float*)d_in[7];
  const float*     att1 = (const float*)d_in[8];
  const float*     bias1= (const float*)d_in[9];
  const float*     Wl2  = (const float*)d_in[10];
  const float*     bl2  = (const float*)d_in[11];
  const float*     Wr2  = (const float*)d_in[12];
  const float*     br2  = (const float*)d_in[13];
  const float*     We2  = (const float*)d_in[14];
  const float*     att2 = (const float*)d_in[15];
  const float*     bias2= (const float*)d_in[16];

  const long long* src = ei;
  const long long* dst = ei + E;

  // ------- workspace layout (layer 2 aliases layer 1 gather buffers) -------
  float* ws = (float*)d_ws;
  size_t off = 0;
  float*    xl1    = ws + off; off += (size_t)N * HC1;   // also xl2
  float*    xr1    = ws + off; off += (size_t)N * HC1;   // also xr2
  float*    h1     = ws + off; off += (size_t)N * HC1;   // layer-1 output
  float*    logits = ws + off; off += (size_t)E * H1;    // logits -> ex (both layers)
  unsigned* mxenc  = (unsigned*)(ws + off); off += (size_t)N * H1;
  float*    mx     = ws + off; off += (size_t)N * H1;
  float*    den    = ws + off; off += (size_t)N * H1;
  float*    xl2 = xl1;
  float*    xr2 = xr1;
  float*    out = (float*)d_out;                         // [N, OUT]

  const int TB  = 256;            // 8 waves per block
  const int WPB = TB / 32;

  // ============================== Layer 1 ==================================
  {
    const int tilesM = N / 16, tilesN = HC1 / 16;
    dim3 g((unsigned)(tilesN * cdiv(tilesM, GEMM_WAVES))), b(TB);
    gemm_bias_wmma<<<g, b, 0, stream>>>(x, Wl1, bl1, xl1, N, HC1, FIN);
    gemm_bias_wmma<<<g, b, 0, stream>>>(x, Wr1, br1, xr1, N, HC1, FIN);
  }
  fill_u32_kernel<<<cdiv((long)N * H1, TB), TB, 0, stream>>>(mxenc, 0u, (long)N * H1);
  edge_logits_kernel<<<cdiv((long)E * H1, WPB), TB, 0, stream>>>(
      xl1, xr1, We1, att1, src, dst, ea, logits, mxenc, E, H1, HID);
  fix_max_kernel<<<cdiv((long)N * H1, TB), TB, 0, stream>>>(mxenc, mx, den, (long)N * H1);
  edge_exp_kernel<<<cdiv((long)E * H1, TB), TB, 0, stream>>>(dst, logits, mx, den, E, H1);
  fill_f32_kernel<<<cdiv((long)N * HC1, TB), TB, 0, stream>>>(h1, 0.f, (long)N * HC1);
  edge_scatter_kernel<<<cdiv((long)E * H1, WPB), TB, 0, stream>>>(
      src, dst, logits, den, xl1, h1, E, H1, HID);
  finalize_elu_kernel<<<cdiv((long)N * HC1, TB), TB, 0, stream>>>(h1, bias1, (long)N * HC1, HC1);

  // ============================== Layer 2 ==================================
  {
    const int tilesM = N / 16, tilesN = OUT / 16;
    dim3 g((unsigned)(tilesN * cdiv(tilesM, GEMM_WAVES))), b(TB);
    gemm_bias_wmma<<<g, b, 0, stream>>>(h1, Wl2, bl2, xl2, N, OUT, HC1);
    gemm_bias_wmma<<<g, b, 0, stream>>>(h1, Wr2, br2, xr2, N, OUT, HC1);
  }
  fill_u32_kernel<<<cdiv((long)N, TB), TB, 0, stream>>>(mxenc, 0u, (long)N);
  edge_logits_kernel<<<cdiv((long)E, WPB), TB, 0, stream>>>(
      xl2, xr2, We2, att2, src, dst, ea, logits, mxenc, E, 1, OUT);
  fix_max_kernel<<<cdiv((long)N, TB), TB, 0, stream>>>(mxenc, mx, den, (long)N);
  edge_exp_kernel<<<cdiv((long)E, TB), TB, 0, stream>>>(dst, logits, mx, den, E, 1);
  fill_f32_kernel<<<cdiv((long)N * OUT, TB), TB, 0, stream>>>(out, 0.f, (long)N * OUT);
  edge_scatter_kernel<<<cdiv((long)E, WPB), TB, 0, stream>>>(
      src, dst, logits, den, xl2, out, E, 1, OUT);
  finalize_elu_kernel<<<cdiv((long)N * OUT, TB), TB, 0, stream>>>(out, bias2, (long)N * OUT, OUT);
}